// HMCModel_61933428417032
// MI455X (gfx1250) — compile-verified
//
#include <hip/hip_runtime.h>
#include <hip/hip_bf16.h>

// ---------------------------------------------------------------------------
// HMC model on gfx1250. Heavy matmuls use V_WMMA_F32_16X16X4_F32 (full fp32
// matrix path, wave32). N = 4096, C = 64, ENC0 = 128, ENC1 = 64, L = 2.
// Big aggregations are split-K x4 for machine utilization (~1024 waves), with
// deterministic fixed-order partial combination.
// ---------------------------------------------------------------------------

#define NN     4096
#define CC     64
#define NCF    (NN * CC)     // floats per [N,C] buffer
#define SPLITK 4
#define KCH    (NN / SPLITK) // 1024 k-values per split

typedef __attribute__((ext_vector_type(2))) float v2f;
typedef __attribute__((ext_vector_type(8))) float v8f;

// f32 WMMA: D(16x16,f32) = A(16x4,f32) * B(4x16,f32) + C
// Lane L: m = L&15, h = L>>4.  A elem e -> A[m][2h+e], B elem e -> B[2h+e][m]
// C/D: VGPR r, lane L -> D[r + 8h][m]
__device__ __forceinline__ v8f wmma_f32(v2f a, v2f b, v8f c) {
    return __builtin_amdgcn_wmma_f32_16x16x4_f32(
        /*neg_a=*/false, a, /*neg_b=*/false, b,
        /*c_mod=*/(short)0, c, /*reuse_a=*/false, /*reuse_b=*/false);
}

// ---------------------------------------------------------------------------
// Small linear: Y = X @ W + b.  X:[N,K] row-major (K=64 or 128), W:[K,64],
// b:[64].  STORE_T=true stores Y transposed as [64,N] so the aggregation
// kernels fetch their B operand as contiguous float2.
// One wave -> 16 rows x 64 cols.  64 blocks x 128 threads (256 waves).
// ---------------------------------------------------------------------------
template <bool STORE_T>
__global__ __launch_bounds__(128)
void lin_kernel(const float* __restrict__ X, const float* __restrict__ W,
                const float* __restrict__ bias, float* __restrict__ Y, int K) {
    const int lane = threadIdx.x & 31;
    const int wave = (blockIdx.x * blockDim.x + threadIdx.x) >> 5;
    const int i0 = wave * 16;
    const int m = lane & 15;
    const int h = lane >> 4;

    v8f acc[4];
#pragma unroll
    for (int jt = 0; jt < 4; ++jt) acc[jt] = (v8f)0.0f;

    for (int k0 = 0; k0 < K; k0 += 4) {
        const int ka = k0 + 2 * h;
        v2f a = *(const v2f*)(X + (size_t)(i0 + m) * K + ka);
#pragma unroll
        for (int jt = 0; jt < 4; ++jt) {
            const int j = jt * 16 + m;
            v2f b;
            b.x = W[(size_t)ka * CC + j];
            b.y = W[(size_t)(ka + 1) * CC + j];
            acc[jt] = wmma_f32(a, b, acc[jt]);
        }
    }

#pragma unroll
    for (int jt = 0; jt < 4; ++jt) {
        const int j = jt * 16 + m;
        const float bj = bias[j];
#pragma unroll
        for (int r = 0; r < 8; ++r) {
            const int row = i0 + r + 8 * h;
            const float v = acc[jt][r] + bj;
            if (STORE_T) Y[(size_t)j * NN + row] = v;      // [64, N]
            else         Y[(size_t)row * CC + j] = v;      // [N, 64]
        }
    }
}

// ---------------------------------------------------------------------------
// Split-K big aggregation (partial): Part[split] = opA(P) @ T [+ opQ(Q) @ U]
// over k in [split*KCH, (split+1)*KCH).
//   P,Q : [N,N] row-major (TRANS* reads the transpose)
//   T,U : [64,N] transposed layout; Part : [SPLITK][N][64] row-major
// One wave -> 16 rows x 64 cols of one split.
// grid = (64, SPLITK), block = 128 (4 row-waves per block).
// Software-pipelined: next k-step's operands are in flight during the WMMAs.
// ---------------------------------------------------------------------------
template <bool TRANSA, bool HASQ, bool TRANSQ>
__global__ __launch_bounds__(128)
void bigmm_partial(const float* __restrict__ P, const float* __restrict__ Tt,
                   const float* __restrict__ Q, const float* __restrict__ Ut,
                   float* __restrict__ Part) {
    const int lane = threadIdx.x & 31;
    const int wrow = (blockIdx.x * blockDim.x + threadIdx.x) >> 5; // 0..255
    const int split = blockIdx.y;                                  // 0..3
    const int i0 = wrow * 16;
    const int m = lane & 15;
    const int h = lane >> 4;
    const int kbeg = split * KCH;
    const int kend = kbeg + KCH;

    v8f acc[4];
#pragma unroll
    for (int jt = 0; jt < 4; ++jt) acc[jt] = (v8f)0.0f;

    // ---- operand load helpers (as lambdas for the pipeline) ----
    auto loadA = [&](int k0, const float* M, bool trans) -> v2f {
        const int ka = k0 + 2 * h;
        v2f a;
        if (!trans) {
            a = *(const v2f*)(M + (size_t)(i0 + m) * NN + ka);
        } else {
            a.x = M[(size_t)ka * NN + i0 + m];
            a.y = M[(size_t)(ka + 1) * NN + i0 + m];
        }
        return a;
    };
    auto loadB = [&](int k0, const float* B, int jt) -> v2f {
        const int j = jt * 16 + m;
        return *(const v2f*)(B + (size_t)j * NN + k0 + 2 * h);
    };

    // ---- prologue: load first k-step ----
    v2f a_c = loadA(kbeg, P, TRANSA);
    v2f aq_c; if (HASQ) aq_c = loadA(kbeg, Q, TRANSQ);
    v2f b_c[4], bu_c[4];
#pragma unroll
    for (int jt = 0; jt < 4; ++jt) {
        b_c[jt] = loadB(kbeg, Tt, jt);
        if (HASQ) bu_c[jt] = loadB(kbeg, Ut, jt);
    }

    // ---- pipelined main loop ----
    for (int k0 = kbeg; k0 < kend - 4; k0 += 4) {
        const int kn = k0 + 4;

        // streaming prefetch, once per 64-float (256 B) chunk (uniform branch)
        if ((kn & 63) == 0) {
            const int kp = kn + 128;
            if (kp < kend) {
                if (!TRANSA)
                    __builtin_prefetch(P + (size_t)(i0 + m) * NN + kp, 0, 1);
                if (HASQ && !TRANSQ)
                    __builtin_prefetch(Q + (size_t)(i0 + m) * NN + kp, 0, 1);
#pragma unroll
                for (int jt = 0; jt < 4; ++jt) {
                    __builtin_prefetch(Tt + (size_t)(jt * 16 + m) * NN + kp, 0, 1);
                    if (HASQ)
                        __builtin_prefetch(Ut + (size_t)(jt * 16 + m) * NN + kp, 0, 1);
                }
            }
        }

        // issue next k-step's loads
        v2f a_n = loadA(kn, P, TRANSA);
        v2f aq_n; if (HASQ) aq_n = loadA(kn, Q, TRANSQ);
        v2f b_n[4], bu_n[4];
#pragma unroll
        for (int jt = 0; jt < 4; ++jt) {
            b_n[jt] = loadB(kn, Tt, jt);
            if (HASQ) bu_n[jt] = loadB(kn, Ut, jt);
        }

        // compute current k-step
#pragma unroll
        for (int jt = 0; jt < 4; ++jt) {
            acc[jt] = wmma_f32(a_c, b_c[jt], acc[jt]);
            if (HASQ) acc[jt] = wmma_f32(aq_c, bu_c[jt], acc[jt]);
        }

        // rotate
        a_c = a_n;
        if (HASQ) aq_c = aq_n;
#pragma unroll
        for (int jt = 0; jt < 4; ++jt) {
            b_c[jt] = b_n[jt];
            if (HASQ) bu_c[jt] = bu_n[jt];
        }
    }
    // ---- epilogue: last k-step ----
#pragma unroll
    for (int jt = 0; jt < 4; ++jt) {
        acc[jt] = wmma_f32(a_c, b_c[jt], acc[jt]);
        if (HASQ) acc[jt] = wmma_f32(aq_c, bu_c[jt], acc[jt]);
    }

    // ---- store partial (no relu / no addend here) ----
    float* out = Part + (size_t)split * NCF;
#pragma unroll
    for (int jt = 0; jt < 4; ++jt) {
        const int j = jt * 16 + m;
#pragma unroll
        for (int r = 0; r < 8; ++r) {
            const int row = i0 + r + 8 * h;
            out[(size_t)row * CC + j] = acc[jt][r];
        }
    }
}

// ---------------------------------------------------------------------------
// Combine split-K partials (fixed order => deterministic):
//   Out = relu(sum_s Part[s] [+ Z]) ;  Z in transposed [64,N] layout.
// ---------------------------------------------------------------------------
template <bool HASZ>
__global__ __launch_bounds__(256)
void combine_kernel(const float* __restrict__ Part, const float* __restrict__ Zt,
                    float* __restrict__ Out) {
    const int idx = blockIdx.x * 256 + threadIdx.x;   // 0 .. NCF-1
    const int row = idx >> 6;
    const int j = idx & 63;
    float v = Part[idx] + Part[NCF + idx] + Part[2 * (size_t)NCF + idx]
            + Part[3 * (size_t)NCF + idx];
    if (HASZ) v += Zt[(size_t)j * NN + row];
    Out[idx] = fmaxf(v, 0.0f);
}

// ---------------------------------------------------------------------------
// Final: out[0] = mean_r(X0@W0+b0) + mean_r(X1@W1+b1) + mean_r(X2@W2+b2)
// (inputs are finite -> nanmean == mean)
// ---------------------------------------------------------------------------
__global__ __launch_bounds__(256)
void reduce_kernel(const float* __restrict__ X0, const float* __restrict__ X1,
                   const float* __restrict__ X2,
                   const float* __restrict__ W0, const float* __restrict__ b0,
                   const float* __restrict__ W1, const float* __restrict__ b1,
                   const float* __restrict__ W2, const float* __restrict__ b2,
                   float* __restrict__ out) {
    __shared__ float sdata[256];
    float s = 0.0f;
    for (int r = threadIdx.x; r < NN; r += 256) {
        const float* x0 = X0 + (size_t)r * CC;
        const float* x1 = X1 + (size_t)r * CC;
        const float* x2 = X2 + (size_t)r * CC;
        float d0 = 0.f, d1 = 0.f, d2 = 0.f;
#pragma unroll 8
        for (int c = 0; c < CC; ++c) {
            d0 += x0[c] * W0[c];
            d1 += x1[c] * W1[c];
            d2 += x2[c] * W2[c];
        }
        s += d0 + d1 + d2;
    }
    sdata[threadIdx.x] = s;
    __syncthreads();
    for (int off = 128; off > 0; off >>= 1) {
        if (threadIdx.x < off) sdata[threadIdx.x] += sdata[threadIdx.x + off];
        __syncthreads();
    }
    if (threadIdx.x == 0)
        out[0] = sdata[0] / (float)NN + b0[0] + b1[0] + b2[0];
}

__global__ void fill0_kernel(float* __restrict__ p, int n) {
    int i = blockIdx.x * blockDim.x + threadIdx.x;
    if (i < n) p[i] = 0.0f;
}

// ---------------------------------------------------------------------------
// Host orchestration
// ---------------------------------------------------------------------------
extern "C" void kernel_launch(void* const* d_in, const int* in_sizes, int n_in,
                              void* d_out, int out_size, void* d_ws, size_t ws_size,
                              hipStream_t stream) {
    (void)in_sizes; (void)n_in; (void)out_size; (void)ws_size;

    const float* x0in  = (const float*)d_in[0];   // [N,128]
    const float* x1in  = (const float*)d_in[1];   // [N,64]
    const float* adj   = (const float*)d_in[2];   // [N,N]
    const float* inc   = (const float*)d_in[3];   // [N,N]
    const float* W0_in = (const float*)d_in[4];   // [128,64]
    const float* b0_in = (const float*)d_in[5];
    const float* W1_in = (const float*)d_in[6];   // [64,64]
    const float* b1_in = (const float*)d_in[7];
    const float* W0_o  = (const float*)d_in[8];   // [64,1]
    const float* b0_o  = (const float*)d_in[9];
    const float* W1_o  = (const float*)d_in[10];
    const float* b1_o  = (const float*)d_in[11];
    const float* W2_o  = (const float*)d_in[12];
    const float* b2_o  = (const float*)d_in[13];

    enum { L1_00 = 0, L1_10, L1_11, L1_21, L2_00, L2_01, L2_11, L2_12, L2_22 };
    const float *LW[9], *LB[9];
    for (int i = 0; i < 9; ++i) {
        LW[i] = (const float*)d_in[14 + 2 * i];
        LB[i] = (const float*)d_in[15 + 2 * i];
    }

    float* S = (float*)d_ws;
    float* S0 = S + 0 * (size_t)NCF;    // x0  [N,64]
    float* S1 = S + 1 * (size_t)NCF;    // x1  [N,64]
    float* S2 = S + 2 * (size_t)NCF;    // x2  [N,64]
    float* SA = S + 3 * (size_t)NCF;    // x0_l1
    float* SB = S + 4 * (size_t)NCF;    // x1_l1
    float* T0 = S + 5 * (size_t)NCF;    // lin scratch, transposed [64,N]
    float* T1 = S + 6 * (size_t)NCF;
    float* T2 = S + 7 * (size_t)NCF;
    float* T3 = S + 8 * (size_t)NCF;
    float* PB = S + 9 * (size_t)NCF;    // split-K partials [SPLITK][N][64]

    const dim3 blkL(128), grdL(NN / 16 / 4);     // 256 waves
    const dim3 blkG(128), grdG(NN / 16 / 4, SPLITK);  // 1024 waves
    const dim3 blkC(256), grdC(NCF / 256);

    // Input embeddings
    lin_kernel<false><<<grdL, blkL, 0, stream>>>(x0in, W0_in, b0_in, S0, 128);
    lin_kernel<false><<<grdL, blkL, 0, stream>>>(x1in, W1_in, b1_in, S1, 64);
    fill0_kernel<<<(NCF + 255) / 256, 256, 0, stream>>>(S2, NCF);

    for (int l = 0; l < 2; ++l) {
        const size_t wo = (size_t)l * CC * CC, bo = (size_t)l * CC;

        // ---- level 1 ----
        lin_kernel<true><<<grdL, blkL, 0, stream>>>(S0, LW[L1_00] + wo, LB[L1_00] + bo, T0, 64);
        lin_kernel<true><<<grdL, blkL, 0, stream>>>(S1, LW[L1_10] + wo, LB[L1_10] + bo, T1, 64);
        // x0_l1 = relu(adj @ t0 + inc^T @ t1)
        bigmm_partial<false, true, true><<<grdG, blkG, 0, stream>>>(adj, T0, inc, T1, PB);
        combine_kernel<false><<<grdC, blkC, 0, stream>>>(PB, nullptr, SA);

        lin_kernel<true><<<grdL, blkL, 0, stream>>>(S1, LW[L1_11] + wo, LB[L1_11] + bo, T2, 64);
        lin_kernel<true><<<grdL, blkL, 0, stream>>>(S2, LW[L1_21] + wo, LB[L1_21] + bo, T3, 64);
        // x1_l1 = relu(inc @ t3 + t2)
        bigmm_partial<false, false, false><<<grdG, blkG, 0, stream>>>(inc, T3, nullptr, nullptr, PB);
        combine_kernel<true><<<grdC, blkC, 0, stream>>>(PB, T2, SB);
        // x2_l1 = x2 (unchanged in S2)

        // ---- level 2 ----
        lin_kernel<true><<<grdL, blkL, 0, stream>>>(SA, LW[L2_00] + wo, LB[L2_00] + bo, T0, 64);
        // x0_n = relu(adj @ u0)
        bigmm_partial<false, false, false><<<grdG, blkG, 0, stream>>>(adj, T0, nullptr, nullptr, PB);
        combine_kernel<false><<<grdC, blkC, 0, stream>>>(PB, nullptr, S0);

        lin_kernel<true><<<grdL, blkL, 0, stream>>>(SA, LW[L2_01] + wo, LB[L2_01] + bo, T1, 64);
        lin_kernel<true><<<grdL, blkL, 0, stream>>>(SB, LW[L2_11] + wo, LB[L2_11] + bo, T2, 64);
        // x1_n = relu(inc^T @ u1 + u2)
        bigmm_partial<true, false, false><<<grdG, blkG, 0, stream>>>(inc, T1, nullptr, nullptr, PB);
        combine_kernel<true><<<grdC, blkC, 0, stream>>>(PB, T2, S1);

        lin_kernel<true><<<grdL, blkL, 0, stream>>>(SB, LW[L2_12] + wo, LB[L2_12] + bo, T0, 64);
        lin_kernel<true><<<grdL, blkL, 0, stream>>>(S2, LW[L2_22] + wo, LB[L2_22] + bo, T1, 64);
        // x2_n = relu(inc @ u3 + u4)
        bigmm_partial<false, false, false><<<grdG, blkG, 0, stream>>>(inc, T0, nullptr, nullptr, PB);
        combine_kernel<true><<<grdC, blkC, 0, stream>>>(PB, T1, S2);
    }

    reduce_kernel<<<1, 256, 0, stream>>>(S0, S1, S2, W0_o, b0_o, W1_o, b1_o,
                                         W2_o, b2_o, (float*)d_out);
}